// NBFNet_4776003633179
// MI455X (gfx1250) — compile-verified
//
#include <hip/hip_runtime.h>
#include <math.h>

#define NN   40000      // N_NODES
#define NE   500000     // N_EDGES
#define NRL  474        // N_REL
#define DIM  32
#define NL   6
#define BB_  4          // BATCH
#define RELF (NRL*DIM)  // 15168
#define KTOT (13*DIM)   // 416
#define ROWS (BB_*NN)   // 160000

typedef float v2f __attribute__((ext_vector_type(2)));
typedef float v8f __attribute__((ext_vector_type(8)));

__device__ __forceinline__ unsigned flipf(float f) {
    unsigned u = __float_as_uint(f);
    return (u & 0x80000000u) ? ~u : (u | 0x80000000u);
}
__device__ __forceinline__ float unflipf(unsigned u) {
    unsigned v = (u & 0x80000000u) ? (u & 0x7fffffffu) : ~u;
    return __uint_as_float(v);
}

// ---------------- one-time kernels ----------------

__global__ void zero_deg_kernel(int* deg) {
    int n = blockIdx.x * blockDim.x + threadIdx.x;
    if (n < NN) deg[n] = 0;
}

__global__ void deg_kernel(const int* dst, int* deg) {
    int e = blockIdx.x * blockDim.x + threadIdx.x;
    if (e < NE) atomicAdd(&deg[dst[e]], 1);
}

// deterministic single-block reduction: smean[0] = mean_n log(deg[n]+1)
__global__ void scale_mean_kernel(const int* deg, float* smean) {
    __shared__ float sh[256];
    float s = 0.f;
    for (int n = threadIdx.x; n < NN; n += 256) s += logf((float)(deg[n] + 1));
    sh[threadIdx.x] = s; __syncthreads();
    for (int o = 128; o > 0; o >>= 1) {
        if (threadIdx.x < o) sh[threadIdx.x] += sh[threadIdx.x + o];
        __syncthreads();
    }
    if (threadIdx.x == 0) smean[0] = sh[0] / (float)NN;
}

__global__ void scale3_kernel(const int* deg, const float* smean, float* scale3) {
    int n = blockIdx.x * blockDim.x + threadIdx.x;
    if (n >= NN) return;
    float s = logf((float)(deg[n] + 1)) / smean[0];
    scale3[n*3+0] = 1.f;
    scale3[n*3+1] = s;
    scale3[n*3+2] = 1.f / fmaxf(s, 0.01f);
}

__global__ void xinit_kernel(float* x, const int* h_index, const int* r_index,
                             const float* query_emb) {
    int t = blockIdx.x * blockDim.x + threadIdx.x;
    if (t >= ROWS * DIM) return;
    int d = t & 31, bn = t >> 5;
    int n = bn % NN, b = bn / NN;
    x[t] = (h_index[b] == n) ? query_emb[r_index[b]*DIM + d] : 0.f;
}

// ---------------- per-layer kernels ----------------

__global__ void rel_kernel(float* rel, const float* query_emb, const int* r_index,
                           const float* rw, const float* rb) {
    int t = blockIdx.x * blockDim.x + threadIdx.x;
    if (t >= BB_ * RELF) return;
    int b = t / RELF, i = t % RELF;
    const float* q = query_emb + r_index[b]*DIM;
    const float* w = rw + (size_t)i * DIM;
    float s = rb[i];
#pragma unroll
    for (int d = 0; d < DIM; ++d) s += q[d] * w[d];
    rel[t] = s;
}

// permW[k'][j]: k'<32 -> lin_w[j][k']; k'=32+128s+f (f=4d+c) -> lin_w[j][32+12d+3c+s]
__global__ void permw_kernel(float* permW, const float* lw) {
    int t = blockIdx.x * blockDim.x + threadIdx.x;
    if (t >= KTOT * DIM) return;
    int kp = t >> 5, j = t & 31;
    int ko;
    if (kp < DIM) ko = kp;
    else { int k4 = kp - DIM; int s = k4 >> 7; int f = k4 & 127;
           int dd = f >> 2; int c = f & 3; ko = DIM + 12*dd + 3*c + s; }
    permW[kp*DIM + j] = lw[(size_t)j*KTOT + ko];
}

// agg slots per (row,d): [0]=sum [1]=maxbits [2]=minbits [3]=sumsq, seeded with boundary
__global__ void agginit_kernel(float* agg, const int* h_index, const int* r_index,
                               const float* query_emb) {
    int t = blockIdx.x * blockDim.x + threadIdx.x;
    if (t >= ROWS * DIM) return;
    int d = t & 31, bn = t >> 5;
    int n = bn % NN, b = bn / NN;
    float bv = (h_index[b] == n) ? query_emb[r_index[b]*DIM + d] : 0.f;
    size_t base = (size_t)t * 4;
    agg[base+0] = bv;
    agg[base+1] = __uint_as_float(flipf(bv));
    agg[base+2] = __uint_as_float(flipf(bv));
    agg[base+3] = bv * bv;
}

__global__ void scatter_kernel(float* agg, const float* x, const float* rel,
                               const int* src, const int* dst, const int* ety) {
    int t = blockIdx.x * blockDim.x + threadIdx.x;
    if (t >= BB_ * NE) return;
    int b = t / NE, e = t % NE;
    const float* xr = x   + (size_t)(b*NN + src[e]) * DIM;
    const float* rr = rel + (size_t)b * RELF + (size_t)ety[e] * DIM;
    float*    ab = agg + (size_t)(b*NN + dst[e]) * DIM * 4;
    unsigned* au = (unsigned*)ab;
#pragma unroll 8
    for (int d = 0; d < DIM; ++d) {
        float m = xr[d] * rr[d];
        atomicAdd(ab + 4*d + 0, m);
        unsigned fm = flipf(m);
        atomicMax(au + 4*d + 1, fm);
        atomicMin(au + 4*d + 2, fm);
        atomicAdd(ab + 4*d + 3, m * m);
    }
}

// in-place: [sum,maxbits,minbits,sumsq] -> [mean,max,min,std]
__global__ void finalize_kernel(float* agg, const int* deg) {
    int t = blockIdx.x * blockDim.x + threadIdx.x;
    if (t >= ROWS * DIM) return;
    int n = (t >> 5) % NN;
    size_t base = (size_t)t * 4;
    float   s0 = agg[base+0];
    unsigned mb = __float_as_uint(agg[base+1]);
    unsigned nb = __float_as_uint(agg[base+2]);
    float   s3 = agg[base+3];
    float df  = (float)(deg[n] + 1);
    float inv = 1.f / df;
    float mean = s0 * inv;
    float var  = s3 * inv - mean * mean;
    agg[base+0] = mean;
    agg[base+1] = unflipf(mb);
    agg[base+2] = unflipf(nb);
    agg[base+3] = sqrtf(fmaxf(var, 1e-6f));
}

// WMMA GEMM: h(row,0:32) = [x | feats*scales] @ permW, then bias+LN+ReLU+residual, x updated in place.
// One wave: 16 rows x 32 cols, K=416 via 104 x V_WMMA_F32_16X16X4_F32 pairs.
__global__ __launch_bounds__(256) void lin_wmma_kernel(float* x, const float* feats,
        const float* scale3, const float* permW,
        const float* lb, const float* g, const float* bb) {
    __shared__ float lds[8][16][32];
    const int wave = threadIdx.x >> 5, lane = threadIdx.x & 31;
    const int m = lane & 15, hf = lane >> 4;
    const int row0 = (blockIdx.x * 8 + wave) * 16;
    const int row = row0 + m;
    const int node = row % NN;
    const float* rowx = x     + (size_t)row * DIM;
    const float* rowf = feats + (size_t)row * 128;
    const float sc0 = scale3[node*3+0], sc1 = scale3[node*3+1], sc2 = scale3[node*3+2];

    v8f acc0 = {}; v8f acc1 = {};
    int kk = 0;
    // k in [0,32): x part
#pragma unroll
    for (int i = 0; i < 8; ++i, ++kk) {
        int kb = kk*4 + hf*2;
        v2f a, b0, b1;
        a.x = rowx[kb]; a.y = rowx[kb+1];
        b0.x = permW[kb*DIM + m];        b0.y = permW[(kb+1)*DIM + m];
        b1.x = permW[kb*DIM + 16 + m];   b1.y = permW[(kb+1)*DIM + 16 + m];
        acc0 = __builtin_amdgcn_wmma_f32_16x16x4_f32(false, a, false, b0, (short)0, acc0, false, false);
        acc1 = __builtin_amdgcn_wmma_f32_16x16x4_f32(false, a, false, b1, (short)0, acc1, false, false);
    }
    // k in [32,416): feats * scale[s]
    for (int s = 0; s < 3; ++s) {
        float scv = (s == 0) ? sc0 : (s == 1) ? sc1 : sc2;
        int f0 = -32 - 128*s;
#pragma unroll 4
        for (int i = 0; i < 32; ++i, ++kk) {
            int kb = kk*4 + hf*2;
            int f = kb + f0;
            v2f a, b0, b1;
            a.x = rowf[f] * scv; a.y = rowf[f+1] * scv;
            b0.x = permW[kb*DIM + m];        b0.y = permW[(kb+1)*DIM + m];
            b1.x = permW[kb*DIM + 16 + m];   b1.y = permW[(kb+1)*DIM + 16 + m];
            acc0 = __builtin_amdgcn_wmma_f32_16x16x4_f32(false, a, false, b0, (short)0, acc0, false, false);
            acc1 = __builtin_amdgcn_wmma_f32_16x16x4_f32(false, a, false, b1, (short)0, acc1, false, false);
        }
    }
    // C/D layout: VGPR i, lane<16 -> M=i ; lane>=16 -> M=i+8 ; N=lane&15
#pragma unroll
    for (int i = 0; i < 8; ++i) {
        int mm = hf ? (i + 8) : i;
        lds[wave][mm][m]      = acc0[i];
        lds[wave][mm][16 + m] = acc1[i];
    }
    __syncthreads();

    if (threadIdx.x < 128) {
        int lr = threadIdx.x, wv = lr >> 4, mm = lr & 15;
        int rrow = (blockIdx.x * 8 + wv) * 16 + mm;
        float h[DIM]; float mu = 0.f;
#pragma unroll
        for (int j = 0; j < DIM; ++j) { h[j] = lds[wv][mm][j] + lb[j]; mu += h[j]; }
        mu *= (1.f / DIM);
        float var = 0.f;
#pragma unroll
        for (int j = 0; j < DIM; ++j) { float dv = h[j] - mu; var += dv * dv; }
        var *= (1.f / DIM);
        float rstd = rsqrtf(var + 1e-5f);
        float* xr = x + (size_t)rrow * DIM;
#pragma unroll
        for (int j = 0; j < DIM; ++j) {
            float v = (h[j] - mu) * rstd * g[j] + bb[j];
            xr[j] = fmaxf(v, 0.f) + xr[j];
        }
    }
}

// final scoring MLP: score = relu([x|query] @ w1 + b1) @ w2 + b2
__global__ __launch_bounds__(256) void mlp_kernel(float* out, const float* x,
        const float* query_emb, const int* r_index,
        const float* w1, const float* b1, const float* w2, const float* b2) {
    __shared__ float w1s[64*64];
    __shared__ float w2s[64], b1s[64];
    for (int i = threadIdx.x; i < 64*64; i += 256) w1s[i] = w1[i];
    if (threadIdx.x < 64) { w2s[threadIdx.x] = w2[threadIdx.x]; b1s[threadIdx.x] = b1[threadIdx.x]; }
    __syncthreads();
    int r = blockIdx.x * 256 + threadIdx.x;
    if (r >= ROWS) return;
    int b = r / NN;
    const float* xr = x + (size_t)r * DIM;
    const float* q = query_emb + r_index[b]*DIM;
    float feat[64];
#pragma unroll
    for (int d = 0; d < DIM; ++d) { feat[d] = xr[d]; feat[DIM+d] = q[d]; }
    float score = b2[0];
    for (int j = 0; j < 64; ++j) {
        float s = b1s[j];
#pragma unroll
        for (int k = 0; k < 64; ++k) s += feat[k] * w1s[k*64 + j];
        score += fmaxf(s, 0.f) * w2s[j];
    }
    out[r] = score;
}

// ---------------- launch ----------------

static inline char* alignp(char* p) {
    return (char*)(((uintptr_t)p + 255) & ~(uintptr_t)255);
}

extern "C" void kernel_launch(void* const* d_in, const int* in_sizes, int n_in,
                              void* d_out, int out_size, void* d_ws, size_t ws_size,
                              hipStream_t stream) {
    const int*   edge_index = (const int*)d_in[0];   // (2, NE)
    const int*   edge_type  = (const int*)d_in[1];
    const int*   h_index    = (const int*)d_in[2];
    const int*   r_index    = (const int*)d_in[3];
    const float* query_emb  = (const float*)d_in[4]; // (NRL, 32)
    const float* rel_w      = (const float*)d_in[5]; // (NL, RELF, 32)
    const float* rel_b      = (const float*)d_in[6]; // (NL, RELF)
    const float* lin_w      = (const float*)d_in[7]; // (NL, 32, 416)
    const float* lin_b      = (const float*)d_in[8]; // (NL, 32)
    const float* ln_g       = (const float*)d_in[9];
    const float* ln_b       = (const float*)d_in[10];
    const float* mlp_w1     = (const float*)d_in[11]; // (64,64)
    const float* mlp_b1     = (const float*)d_in[12];
    const float* mlp_w2     = (const float*)d_in[13]; // (64,1)
    const float* mlp_b2     = (const float*)d_in[14];
    float* out = (float*)d_out;

    const int* src = edge_index;
    const int* dst = edge_index + NE;

    // workspace carve (~103.5 MB)
    char* p = (char*)d_ws;
    float* agg    = (float*)p; p = alignp(p + (size_t)ROWS*DIM*4*sizeof(float)); // 81.92 MB
    float* x      = (float*)p; p = alignp(p + (size_t)ROWS*DIM*sizeof(float));   // 20.48 MB
    float* rel    = (float*)p; p = alignp(p + (size_t)BB_*RELF*sizeof(float));
    float* permW  = (float*)p; p = alignp(p + (size_t)KTOT*DIM*sizeof(float));
    int*   deg    = (int*)p;   p = alignp(p + (size_t)NN*sizeof(int));
    float* scale3 = (float*)p; p = alignp(p + (size_t)NN*3*sizeof(float));
    float* smean  = (float*)p;

    const int T = 256;
    // layer-invariant: degree, log-degree mean, scales, x = boundary
    zero_deg_kernel<<<(NN + T - 1)/T, T, 0, stream>>>(deg);
    deg_kernel<<<(NE + T - 1)/T, T, 0, stream>>>(dst, deg);
    scale_mean_kernel<<<1, T, 0, stream>>>(deg, smean);
    scale3_kernel<<<(NN + T - 1)/T, T, 0, stream>>>(deg, smean, scale3);
    xinit_kernel<<<(ROWS*DIM + T - 1)/T, T, 0, stream>>>(x, h_index, r_index, query_emb);

    for (int l = 0; l < NL; ++l) {
        rel_kernel<<<(BB_*RELF + T - 1)/T, T, 0, stream>>>(
            rel, query_emb, r_index, rel_w + (size_t)l*RELF*DIM, rel_b + (size_t)l*RELF);
        permw_kernel<<<(KTOT*DIM + T - 1)/T, T, 0, stream>>>(
            permW, lin_w + (size_t)l*DIM*KTOT);
        agginit_kernel<<<(ROWS*DIM + T - 1)/T, T, 0, stream>>>(
            agg, h_index, r_index, query_emb);
        scatter_kernel<<<(BB_*NE + T - 1)/T, T, 0, stream>>>(
            agg, x, rel, src, dst, edge_type);
        finalize_kernel<<<(ROWS*DIM + T - 1)/T, T, 0, stream>>>(agg, deg);
        lin_wmma_kernel<<<ROWS/128, T, 0, stream>>>(
            x, agg, scale3, permW, lin_b + l*DIM, ln_g + l*DIM, ln_b + l*DIM);
    }

    mlp_kernel<<<(ROWS + T - 1)/T, T, 0, stream>>>(
        out, x, query_emb, r_index, mlp_w1, mlp_b1, mlp_w2, mlp_b2);
}